// DwTNL_module_1649267441974
// MI455X (gfx1250) — compile-verified
//
#include <hip/hip_runtime.h>
#include <hip/hip_bf16.h>
#include <cmath>

// ---------------------------------------------------------------------------
// B=8, C=256, T=16, H=W=28 -> CR=128, HW=784, N=T*HW=12544, k_top=196
// All matrix math in fp32 via V_WMMA_F32_16X16X4_F32 (fp32 in/acc) to match
// the fp32 reference. Top-k via exact 4-pass radix select + 256-elem bitonic
// (instead of a full 1024-elem sort: ~18x less LDS/VALU work on the
// dominant kernel; HBM roofline for the whole op is only ~26us @ 23.3 TB/s).
// ---------------------------------------------------------------------------

typedef float v2f __attribute__((ext_vector_type(2)));
typedef float v8f __attribute__((ext_vector_type(8)));

__device__ __forceinline__ v8f wmma_f32_4(v2f a, v2f b, v8f c) {
    return __builtin_amdgcn_wmma_f32_16x16x4_f32(
        false, a, false, b, (short)0, c, false, false);
}

#define NTOT 12544   // T*H*W
#define CIN  256
#define CR   128
#define HW   784
#define TT   16
#define KTOP 196

// ---------------------------------------------------------------------------
// Kernel 1: fused Q/K/V 1x1-conv projections, register-blocked 1Mx4N tiles
// per wave (one weight b64 load feeds 4 WMMAs per K-step).
// wave id = (((b*3 + q)*8 + mt)*196 + ng), ng = group of 4 N-tiles (64 cols).
// ---------------------------------------------------------------------------
__global__ void qkv_proj_kernel(const float* __restrict__ x,
                                const float* __restrict__ Wq, const float* __restrict__ bq,
                                const float* __restrict__ Wk, const float* __restrict__ bk,
                                const float* __restrict__ Wv, const float* __restrict__ bv,
                                float* __restrict__ Q, float* __restrict__ K,
                                float* __restrict__ V) {
    int wave = blockIdx.x * (blockDim.x >> 5) + (threadIdx.x >> 5);
    int lane = threadIdx.x & 31;
    int ng = wave % 196; wave /= 196;
    int mt = wave % 8;   wave /= 8;
    int q  = wave % 3;   wave /= 3;
    int b  = wave;                       // 0..7

    const float* W    = (q == 0) ? Wq : (q == 1) ? Wk : Wv;
    const float* bias = (q == 0) ? bq : (q == 1) ? bk : bv;
    float*       out  = (q == 0) ? Q  : (q == 1) ? K  : V;

    const int lm  = lane & 15;
    const int kof = (lane >> 4) << 1;    // 0 or 2
    const int m0 = mt * 16, n0 = ng * 64;
    const float* xb = x + (size_t)b * CIN * NTOT;

    v8f acc[4];
#pragma unroll
    for (int j = 0; j < 4; ++j) acc[j] = (v8f){0.f,0.f,0.f,0.f,0.f,0.f,0.f,0.f};

    for (int k0 = 0; k0 < CIN; k0 += 4) {
        v2f a;
        const float* wrow = W + (size_t)(m0 + lm) * CIN + k0 + kof;
        a.x = wrow[0];
        a.y = wrow[1];
        const float* xp = xb + (size_t)(k0 + kof) * NTOT + n0 + lm;
#pragma unroll
        for (int j = 0; j < 4; ++j) {
            v2f bb;
            bb.x = xp[j * 16];
            bb.y = xp[j * 16 + NTOT];
            acc[j] = wmma_f32_4(a, bb, acc[j]);
        }
    }

    const int mbase = m0 + (lane >> 4) * 8;
#pragma unroll
    for (int i = 0; i < 8; ++i) {
        int m = mbase + i;
        float bs = bias[m];
        size_t rb = ((size_t)(b * CR + m)) * NTOT + n0 + lm;
#pragma unroll
        for (int j = 0; j < 4; ++j)
            out[rb + j * 16] = acc[j][i] + bs;
    }
}

// ---------------------------------------------------------------------------
// Kernel 2: exact sorted top-196 of each 784-row.
//   1) monotonic float->uint keys (order-preserving)
//   2) 4-pass 8-bit MSB radix select -> exact 196th-largest key T and
//      m = #{keys > T} (<=195), handling ties exactly
//   3) gather keys > T, pad to 256 with T, bitonic-sort 256 desc
//   4) first 196 decoded = jax.lax.top_k values (sorted descending)
// One row per 256-thread block. Blocks [0,16384)=Q rows, rest=K rows.
// ---------------------------------------------------------------------------
__global__ void topk_select_kernel(const float* __restrict__ Q,
                                   const float* __restrict__ K,
                                   float* __restrict__ Qtk,
                                   float* __restrict__ Ktk) {
    __shared__ unsigned keys[HW];
    __shared__ unsigned hist[256];
    __shared__ unsigned sufx[256];
    __shared__ unsigned cand[256];
    __shared__ unsigned sh_d, sh_cntgt, sh_pos, sh_prefix, sh_r;

    int r0 = blockIdx.x;
    const float* src;
    float* dst;
    if (r0 < 16384) { src = Q + (size_t)r0 * HW;               dst = Qtk + (size_t)r0 * KTOP; }
    else            { int rr = r0 - 16384;
                      src = K + (size_t)rr * HW;               dst = Ktk + (size_t)rr * KTOP; }
    const int tid = threadIdx.x;

    for (int i = tid; i < HW; i += 256) {
        unsigned f = __float_as_uint(src[i]);
        keys[i] = f ^ ((f & 0x80000000u) ? 0xFFFFFFFFu : 0x80000000u);
    }
    if (tid == 0) { sh_prefix = 0u; sh_r = KTOP; sh_pos = 0u; }
    __syncthreads();

    for (int p = 3; p >= 0; --p) {
        hist[tid] = 0u;
        __syncthreads();
        unsigned prefix = sh_prefix;
        int shift = p * 8;
        for (int i = tid; i < HW; i += 256) {
            unsigned k = keys[i];
            if ((p == 3) || ((k >> (shift + 8)) == prefix))
                atomicAdd(&hist[(k >> shift) & 0xFFu], 1u);
        }
        __syncthreads();
        // inclusive suffix scan (Hillis-Steele)
        sufx[tid] = hist[tid];
        __syncthreads();
        for (int s = 1; s < 256; s <<= 1) {
            unsigned v = (tid + s < 256) ? sufx[tid + s] : 0u;
            __syncthreads();
            sufx[tid] += v;
            __syncthreads();
        }
        unsigned r = sh_r;
        unsigned above = (tid < 255) ? sufx[tid + 1] : 0u;
        if (sufx[tid] >= r && above < r) {   // exactly one tid (sufx non-increasing)
            sh_d = (unsigned)tid;
            sh_cntgt = above;
        }
        __syncthreads();
        if (tid == 0) {
            sh_r = sh_r - sh_cntgt;
            sh_prefix = (sh_prefix << 8) | sh_d;
        }
        __syncthreads();
    }
    const unsigned T = sh_prefix;            // exact 196th-largest key

    cand[tid] = T;                           // pad with T (covers ties exactly)
    __syncthreads();
    for (int i = tid; i < HW; i += 256) {
        unsigned k = keys[i];
        if (k > T) {
            unsigned idx = atomicAdd(&sh_pos, 1u);   // idx <= 194 < 256 guaranteed
            cand[idx] = k;
        }
    }
    __syncthreads();

    // bitonic sort of 256 keys, descending
    for (int kk = 2; kk <= 256; kk <<= 1) {
        for (int j = kk >> 1; j > 0; j >>= 1) {
            int i = tid;
            int ixj = i ^ j;
            if (ixj > i) {
                unsigned a = cand[i], b = cand[ixj];
                bool desc = ((i & kk) == 0);
                if (desc ? (a < b) : (a > b)) { cand[i] = b; cand[ixj] = a; }
            }
            __syncthreads();
        }
    }

    if (tid < KTOP) {
        unsigned u = cand[tid];
        unsigned f = u ^ ((u & 0x80000000u) ? 0x80000000u : 0xFFFFFFFFu);
        dst[tid] = __uint_as_float(f);
    }
}

// ---------------------------------------------------------------------------
// Kernel 3: per-head attention (one head of 1024 per 256-thread block).
//   corr[16,16] = Qtk[16,196] x Ktk[16,196]^T   (wave 0, 49 WMMAs)
//   row softmax (threads 0-15)
//   Y[16,784]   = attn[16,16] x V[16,784]       (8 waves x 49 N-tiles)
// ---------------------------------------------------------------------------
__global__ void attn_kernel(const float* __restrict__ Qtk,
                            const float* __restrict__ Ktk,
                            const float* __restrict__ V,
                            float* __restrict__ Y) {
    __shared__ float sQ[TT * KTOP];
    __shared__ float sK[TT * KTOP];
    __shared__ float sAttn[TT][TT];

    const int n   = blockIdx.x;
    const int tid = threadIdx.x;
    const int lane = tid & 31, wid = tid >> 5;
    const int lm = lane & 15;
    const int kof = (lane >> 4) << 1;

    const float* qb = Qtk + (size_t)n * TT * KTOP;
    const float* kb = Ktk + (size_t)n * TT * KTOP;
    for (int i = tid; i < TT * KTOP; i += 256) { sQ[i] = qb[i]; sK[i] = kb[i]; }
    __syncthreads();

    if (wid == 0) {
        v8f c = {0.f,0.f,0.f,0.f,0.f,0.f,0.f,0.f};
        for (int k0 = 0; k0 < KTOP; k0 += 4) {       // 196 = 4*49
            v2f a, bb;
            a.x  = sQ[lm * KTOP + k0 + kof];
            a.y  = sQ[lm * KTOP + k0 + kof + 1];
            bb.x = sK[lm * KTOP + k0 + kof];         // B[k][s] = Ktk[s][k]
            bb.y = sK[lm * KTOP + k0 + kof + 1];
            c = wmma_f32_4(a, bb, c);
        }
        int mbase = (lane >> 4) * 8;
#pragma unroll
        for (int i = 0; i < 8; ++i) sAttn[mbase + i][lm] = c[i];
    }
    __syncthreads();

    if (tid < TT) {
        float mx = -3.4028235e38f;
        for (int sI = 0; sI < TT; ++sI) mx = fmaxf(mx, sAttn[tid][sI]);
        float e[TT]; float sum = 0.f;
        for (int sI = 0; sI < TT; ++sI) { e[sI] = expf(sAttn[tid][sI] - mx); sum += e[sI]; }
        float inv = 1.0f / sum;
        for (int sI = 0; sI < TT; ++sI) sAttn[tid][sI] = e[sI] * inv;
    }
    __syncthreads();

    const float* vb = V + (size_t)n * TT * HW;
    float*       yb = Y + (size_t)n * TT * HW;
    for (int nt = wid; nt < 49; nt += 8) {
        int n0 = nt * 16;
        v8f c = {0.f,0.f,0.f,0.f,0.f,0.f,0.f,0.f};
#pragma unroll
        for (int kt = 0; kt < 4; ++kt) {
            int k0 = kt * 4;
            v2f a, bb;
            a.x  = sAttn[lm][k0 + kof];
            a.y  = sAttn[lm][k0 + kof + 1];
            bb.x = vb[(size_t)(k0 + kof)     * HW + n0 + lm];
            bb.y = vb[(size_t)(k0 + kof + 1) * HW + n0 + lm];
            c = wmma_f32_4(a, bb, c);
        }
        int mbase = (lane >> 4) * 8;
#pragma unroll
        for (int i = 0; i < 8; ++i)
            yb[(size_t)(mbase + i) * HW + n0 + lm] = c[i];
    }
}

// ---------------------------------------------------------------------------
// Kernel 4: reconstruct [256,128]x[128,12544] per batch, register-blocked
// 1Mx4N, fused bias + eval-BatchNorm + residual.
// wave id = ((b*16 + mt)*196 + ng)
// ---------------------------------------------------------------------------
__global__ void recon_kernel(const float* __restrict__ Y,
                             const float* __restrict__ Wr, const float* __restrict__ br,
                             const float* __restrict__ gamma, const float* __restrict__ beta,
                             const float* __restrict__ bn_mean, const float* __restrict__ bn_var,
                             const float* __restrict__ x,
                             float* __restrict__ out) {
    int wave = blockIdx.x * (blockDim.x >> 5) + (threadIdx.x >> 5);
    int lane = threadIdx.x & 31;
    int ng = wave % 196; wave /= 196;
    int mt = wave % 16;  wave /= 16;
    int b  = wave;                        // 0..7

    const int lm  = lane & 15;
    const int kof = (lane >> 4) << 1;
    const int m0 = mt * 16, n0 = ng * 64;
    const float* yb = Y + (size_t)b * CR * NTOT;

    v8f acc[4];
#pragma unroll
    for (int j = 0; j < 4; ++j) acc[j] = (v8f){0.f,0.f,0.f,0.f,0.f,0.f,0.f,0.f};

    for (int k0 = 0; k0 < CR; k0 += 4) {
        v2f a;
        const float* wrow = Wr + (size_t)(m0 + lm) * CR + k0 + kof;
        a.x = wrow[0];
        a.y = wrow[1];
        const float* yp = yb + (size_t)(k0 + kof) * NTOT + n0 + lm;
#pragma unroll
        for (int j = 0; j < 4; ++j) {
            v2f bb;
            bb.x = yp[j * 16];
            bb.y = yp[j * 16 + NTOT];
            acc[j] = wmma_f32_4(a, bb, acc[j]);
        }
    }

    const int mbase = m0 + (lane >> 4) * 8;
#pragma unroll
    for (int i = 0; i < 8; ++i) {
        int m = mbase + i;
        float sc = gamma[m] / sqrtf(bn_var[m] + 1e-5f);
        float sh = beta[m] - bn_mean[m] * sc;
        float bs = br[m];
        size_t rb = ((size_t)(b * CIN + m)) * NTOT + n0 + lm;
#pragma unroll
        for (int j = 0; j < 4; ++j) {
            size_t idx = rb + j * 16;
            out[idx] = (acc[j][i] + bs) * sc + sh + x[idx];
        }
    }
}

// ---------------------------------------------------------------------------
// Launch. Inputs (setup_inputs order):
//  0:x 1:Wq 2:bq 3:Wk 4:bk 5:Wv 6:bv 7:Wr 8:br 9:gamma 10:beta 11:bn_mean 12:bn_var
// Workspace: Q | K | V (each 1024*16*784 f32) | Qtk | Ktk (each 1024*16*196)
// Y reuses Q's slot (Q dead after top-k).  Total ~180 MB.
// ---------------------------------------------------------------------------
extern "C" void kernel_launch(void* const* d_in, const int* in_sizes, int n_in,
                              void* d_out, int out_size, void* d_ws, size_t ws_size,
                              hipStream_t stream) {
    const float* x      = (const float*)d_in[0];
    const float* Wq     = (const float*)d_in[1];
    const float* bq     = (const float*)d_in[2];
    const float* Wk     = (const float*)d_in[3];
    const float* bk     = (const float*)d_in[4];
    const float* Wv     = (const float*)d_in[5];
    const float* bv     = (const float*)d_in[6];
    const float* Wr     = (const float*)d_in[7];
    const float* br     = (const float*)d_in[8];
    const float* gamma  = (const float*)d_in[9];
    const float* beta   = (const float*)d_in[10];
    const float* bmean  = (const float*)d_in[11];
    const float* bvar   = (const float*)d_in[12];
    float* out = (float*)d_out;

    float* ws = (float*)d_ws;
    const size_t QKV_ELEMS = (size_t)1024 * TT * HW;
    const size_t TK_ELEMS  = (size_t)1024 * TT * KTOP;
    float* Q   = ws;
    float* K   = Q + QKV_ELEMS;
    float* V   = K + QKV_ELEMS;
    float* Qtk = V + QKV_ELEMS;
    float* Ktk = Qtk + TK_ELEMS;
    float* Y   = Q;   // reuse

    // 8 batches * 3 mats * 8 Mtiles * 196 N-groups = 37,632 waves / 8 per block
    qkv_proj_kernel<<<37632 / 8, 256, 0, stream>>>(x, Wq, bq, Wk, bk, Wv, bv, Q, K, V);
    // 16384 Q rows + 16384 K rows
    topk_select_kernel<<<32768, 256, 0, stream>>>(Q, K, Qtk, Ktk);
    // 1024 heads
    attn_kernel<<<1024, 256, 0, stream>>>(Qtk, Ktk, V, Y);
    // 8 batches * 16 Mtiles * 196 N-groups = 25,088 waves / 8 per block
    recon_kernel<<<25088 / 8, 256, 0, stream>>>(Y, Wr, br, gamma, beta, bmean, bvar, x, out);
}